// custom_PHNN_31404800869100
// MI455X (gfx1250) — compile-verified
//
#include <hip/hip_runtime.h>
#include <hip/hip_bf16.h>
#include <math.h>

// ---------------------------------------------------------------------------
// PHNN rollout for MI455X (gfx1250), fp32 end-to-end using V_WMMA_F32_16X16X4_F32.
// B=256, NA=NB=8, SIG=16, XC=64, T=8, HID=256, DT=0.01
// ---------------------------------------------------------------------------

typedef float v2f __attribute__((ext_vector_type(2)));
typedef float v8f __attribute__((ext_vector_type(8)));

#define LDP 68          // padded LDS leading dim (64 + 4) to dodge bank conflicts

#define MODE_LIN       0   // C = acc + bias
#define MODE_TANH      1   // C = tanh(acc + bias)
#define MODE_DTANH_W3  2   // C = aux[col] * (1 - tanh(acc+bias)^2)
#define MODE_MUL_DTANH 3   // C = acc * (1 - aux[row,col]^2)       (no bias)
#define MODE_ADD       4   // C = acc + bias + aux[row,col]

// Generic fp32 WMMA GEMM: C[M,N] = epilogue(A[M,K] @ op(B) + bias)
// A row-major (lda), B row-major (ldb); transB=1 means use B^T (B is [N,K]).
// Block: 128 threads (4 waves), 64x64 C tile; each wave does 32x32 via 2x2
// tiles of v_wmma_f32_16x16x4_f32. All of M,N divisible by 64; K by 64.
__global__ __launch_bounds__(128) void wmma_gemm_f32(
    const float* __restrict__ A, int lda,
    const float* __restrict__ B, int ldb, int transB,
    const float* __restrict__ bias, const float* __restrict__ aux,
    float* __restrict__ C, int ldc, int K, int mode)
{
    __shared__ float As[64 * LDP];   // As[m][k]
    __shared__ float Bs[64 * LDP];   // Bs[n][k]  (B stored transposed)

    const int tid  = threadIdx.x;
    const int wave = tid >> 5;
    const int lane = tid & 31;
    const int half = lane >> 4;      // 0: lanes 0-15, 1: lanes 16-31
    const int lr   = lane & 15;
    const int wm   = (wave >> 1) * 32;
    const int wn   = (wave & 1) * 32;
    const int bm   = blockIdx.y * 64;
    const int bn   = blockIdx.x * 64;

    v8f acc[2][2];
    acc[0][0] = 0.0f; acc[0][1] = 0.0f; acc[1][0] = 0.0f; acc[1][1] = 0.0f;

    for (int k0 = 0; k0 < K; k0 += 64) {
        // --- stage A tile [64 x 64] via float4 (8 vec4 per thread) ---
        #pragma unroll
        for (int it = 0; it < 8; ++it) {
            int idx = it * 128 + tid;          // 0..1023
            int r = idx >> 4, q = idx & 15;
            float4 v = *(const float4*)&A[(size_t)(bm + r) * lda + k0 + q * 4];
            *(float4*)&As[r * LDP + q * 4] = v;
        }
        // --- stage B tile transposed into Bs[n][k] ---
        if (transB) {
            #pragma unroll
            for (int it = 0; it < 8; ++it) {
                int idx = it * 128 + tid;
                int r = idx >> 4, q = idx & 15;
                float4 v = *(const float4*)&B[(size_t)(bn + r) * ldb + k0 + q * 4];
                *(float4*)&Bs[r * LDP + q * 4] = v;
            }
        } else {
            #pragma unroll
            for (int it = 0; it < 32; ++it) {
                int idx = it * 128 + tid;      // 0..4095
                int kk = idx >> 6, n = idx & 63;
                Bs[n * LDP + kk] = B[(size_t)(k0 + kk) * ldb + bn + n];
            }
        }
        __syncthreads();

        // --- 16 WMMA sweeps over this K-stage (K step = 4) ---
        #pragma unroll
        for (int kt = 0; kt < 64; kt += 4) {
            // A frag 16x4: lanes 0-15 hold K=kt..kt+1, lanes 16-31 K=kt+2..kt+3
            v2f a0 = *(const v2f*)&As[(wm +      lr) * LDP + kt + 2 * half];
            v2f a1 = *(const v2f*)&As[(wm + 16 + lr) * LDP + kt + 2 * half];
            // B frag 4x16 (from transposed LDS tile): symmetric layout
            v2f b0 = *(const v2f*)&Bs[(wn +      lr) * LDP + kt + 2 * half];
            v2f b1 = *(const v2f*)&Bs[(wn + 16 + lr) * LDP + kt + 2 * half];
            acc[0][0] = __builtin_amdgcn_wmma_f32_16x16x4_f32(false, a0, false, b0, (short)0, acc[0][0], false, false);
            acc[0][1] = __builtin_amdgcn_wmma_f32_16x16x4_f32(false, a0, false, b1, (short)0, acc[0][1], false, false);
            acc[1][0] = __builtin_amdgcn_wmma_f32_16x16x4_f32(false, a1, false, b0, (short)0, acc[1][0], false, false);
            acc[1][1] = __builtin_amdgcn_wmma_f32_16x16x4_f32(false, a1, false, b1, (short)0, acc[1][1], false, false);
        }
        __syncthreads();
    }

    // --- epilogue: D layout = lanes 0-15: (M=r, N=lr); lanes 16-31: (M=r+8, N=lr)
    #pragma unroll
    for (int tm = 0; tm < 2; ++tm) {
        #pragma unroll
        for (int tn = 0; tn < 2; ++tn) {
            int colg = bn + wn + tn * 16 + lr;
            float bv = bias ? bias[colg] : 0.0f;
            #pragma unroll
            for (int r = 0; r < 8; ++r) {
                int rowg = bm + wm + tm * 16 + r + 8 * half;
                float v = acc[tm][tn][r];
                if (mode == MODE_LIN) {
                    v += bv;
                } else if (mode == MODE_TANH) {
                    v = tanhf(v + bv);
                } else if (mode == MODE_DTANH_W3) {
                    float t = tanhf(v + bv);
                    v = aux[colg] * (1.0f - t * t);
                } else if (mode == MODE_MUL_DTANH) {
                    float m = aux[(size_t)rowg * ldc + colg];
                    v = v * (1.0f - m * m);
                } else { // MODE_ADD
                    v += bv + aux[(size_t)rowg * ldc + colg];
                }
                C[(size_t)rowg * ldc + colg] = v;
            }
        }
    }
}

// Per-sample dynamics: f = (A - A^T) dH - L (L^T dH) + G u ; optional yhat = G^T dH
__global__ __launch_bounds__(64) void dyn_kernel(
    const float* __restrict__ Amat, const float* __restrict__ Lmat,
    const float* __restrict__ Gmat, const float* __restrict__ dHdx,
    const float* __restrict__ u, int ustride,
    float* __restrict__ kout, float* __restrict__ yhat)
{
    const int b = blockIdx.x;
    const int i = threadIdx.x;
    __shared__ float sdH[64], st[64], sU[16];
    sdH[i] = dHdx[b * 64 + i];
    if (i < 16) sU[i] = u[(size_t)b * ustride + i];
    __syncthreads();

    const float* Ab = Amat + (size_t)b * 4096;
    const float* Lb = Lmat + (size_t)b * 4096;
    const float* Gb = Gmat + (size_t)b * 1024;

    float t = 0.0f;                       // t_j = sum_i L[i,j] dH_i
    #pragma unroll 8
    for (int r = 0; r < 64; ++r) t += Lb[r * 64 + i] * sdH[r];
    st[i] = t;
    __syncthreads();

    float f = 0.0f;
    #pragma unroll 8
    for (int j = 0; j < 64; ++j)
        f += (Ab[i * 64 + j] - Ab[j * 64 + i]) * sdH[j] - Lb[i * 64 + j] * st[j];
    #pragma unroll
    for (int j = 0; j < 16; ++j) f += Gb[i * 16 + j] * sU[j];
    kout[b * 64 + i] = f;

    if (yhat != nullptr && i < 16) {
        float y = 0.0f;
        #pragma unroll 8
        for (int r = 0; r < 64; ++r) y += Gb[r * 16 + i] * sdH[r];
        yhat[(size_t)b * 128 + i] = y;    // stride T*SIG = 128, base pre-offset by t*16
    }
}

__global__ void axpy_kernel(const float* __restrict__ x, const float* __restrict__ k,
                            float c, float* __restrict__ out, int n) {
    int i = blockIdx.x * blockDim.x + threadIdx.x;
    if (i < n) out[i] = x[i] + c * k[i];
}

__global__ void rk4_combine_kernel(const float* __restrict__ x,
                                   const float* __restrict__ k1, const float* __restrict__ k2,
                                   const float* __restrict__ k3, const float* __restrict__ k4,
                                   float h6, float* __restrict__ out, int n) {
    int i = blockIdx.x * blockDim.x + threadIdx.x;
    if (i < n) out[i] = x[i] + h6 * (k1[i] + 2.0f * (k2[i] + k3[i]) + k4[i]);
}

__global__ void pack_kernel(const float* __restrict__ u_past, const float* __restrict__ y_past,
                            float* __restrict__ data, int n) {
    int i = blockIdx.x * blockDim.x + threadIdx.x;
    if (i < n) {
        int b = i >> 8, c = i & 255;
        data[i] = (c < 128) ? u_past[b * 128 + c] : y_past[b * 128 + (c - 128)];
    }
}

// ---------------------------------------------------------------------------
static inline void gemm(hipStream_t s, const float* A, int lda,
                        const float* B, int ldb, int transB,
                        const float* bias, const float* aux,
                        float* C, int ldc, int M, int N, int K, int mode) {
    dim3 grid(N / 64, M / 64);
    wmma_gemm_f32<<<grid, 128, 0, s>>>(A, lda, B, ldb, transB, bias, aux, C, ldc, K, mode);
}

extern "C" void kernel_launch(void* const* d_in, const int* in_sizes, int n_in,
                              void* d_out, int out_size, void* d_ws, size_t ws_size,
                              hipStream_t stream) {
    (void)in_sizes; (void)n_in; (void)out_size; (void)ws_size;
    const float* u_past   = (const float*)d_in[0];
    const float* y_past   = (const float*)d_in[1];
    const float* u_future = (const float*)d_in[2];
    const float* enc_wsk  = (const float*)d_in[3];
    const float* enc_bsk  = (const float*)d_in[4];
    const float* enc_w1   = (const float*)d_in[5];
    const float* enc_b1   = (const float*)d_in[6];
    const float* enc_w2   = (const float*)d_in[7];
    const float* enc_b2   = (const float*)d_in[8];
    const float* enc_w3   = (const float*)d_in[9];
    const float* enc_b3   = (const float*)d_in[10];
    const float* H_w1 = (const float*)d_in[11];
    const float* H_b1 = (const float*)d_in[12];
    const float* H_w2 = (const float*)d_in[13];
    const float* H_b2 = (const float*)d_in[14];
    const float* H_w3 = (const float*)d_in[15];   // [256,1] -> vector[256]
    const float* J_w1 = (const float*)d_in[17];
    const float* J_b1 = (const float*)d_in[18];
    const float* J_w2 = (const float*)d_in[19];
    const float* J_b2 = (const float*)d_in[20];
    const float* R_w1 = (const float*)d_in[21];
    const float* R_b1 = (const float*)d_in[22];
    const float* R_w2 = (const float*)d_in[23];
    const float* R_b2 = (const float*)d_in[24];
    const float* G_w1 = (const float*)d_in[25];
    const float* G_b1 = (const float*)d_in[26];
    const float* G_w2 = (const float*)d_in[27];
    const float* G_b2 = (const float*)d_in[28];

    const float DT = 0.01f;
    float* ws = (float*)d_ws;
    size_t off = 0;
    auto alloc = [&](size_t n) { float* p = ws + off; off += n; return p; };

    float* xA    = alloc(16384);  // 256*64
    float* xB    = alloc(16384);
    float* xa    = alloc(16384);
    float* k1    = alloc(16384);
    float* k2    = alloc(16384);
    float* k3    = alloc(16384);
    float* k4    = alloc(16384);
    float* xsk   = alloc(16384);
    float* dHdx  = alloc(16384);
    float* data  = alloc(65536);  // 256*256
    float* ench1 = alloc(65536);
    float* ench2 = alloc(65536);
    float* h1H   = alloc(65536);
    float* h1J   = alloc(65536);
    float* h1R   = alloc(65536);
    float* h1G   = alloc(65536);
    float* dh2   = alloc(65536);
    float* dh1   = alloc(65536);
    float* Abuf  = alloc(1048576); // 256*4096
    float* Lbuf  = alloc(1048576);
    float* Gbuf  = alloc(262144);  // 256*1024

    // ---------------- encoder ----------------
    pack_kernel<<<256, 256, 0, stream>>>(u_past, y_past, data, 65536);
    gemm(stream, data,  256, enc_w1, 256, 0, enc_b1, nullptr, ench1, 256, 256, 256, 256, MODE_TANH);
    gemm(stream, ench1, 256, enc_w2, 256, 0, enc_b2, nullptr, ench2, 256, 256, 256, 256, MODE_TANH);
    gemm(stream, data,  256, enc_wsk, 64, 0, enc_bsk, nullptr, xsk,  64,  256, 64,  256, MODE_LIN);
    gemm(stream, ench2, 256, enc_w3,  64, 0, enc_b3,  xsk,     xA,   64,  256, 64,  256, MODE_ADD);

    // get_matrices(xs) -> dHdx, Abuf, Lbuf, Gbuf
    auto eval = [&](const float* xs) {
        // Hamiltonian forward + gradient
        gemm(stream, xs,  64,  H_w1, 256, 0, H_b1, nullptr, h1H, 256, 256, 256, 64,  MODE_TANH);
        gemm(stream, h1H, 256, H_w2, 256, 0, H_b2, H_w3,    dh2, 256, 256, 256, 256, MODE_DTANH_W3);
        gemm(stream, dh2, 256, H_w2, 256, 1, nullptr, h1H,  dh1, 256, 256, 256, 256, MODE_MUL_DTANH);
        gemm(stream, dh1, 256, H_w1, 256, 1, nullptr, nullptr, dHdx, 64, 256, 64, 256, MODE_LIN);
        // J / R / G heads
        gemm(stream, xs,  64,  J_w1, 256,  0, J_b1, nullptr, h1J,  256,  256, 256,  64,  MODE_TANH);
        gemm(stream, h1J, 256, J_w2, 4096, 0, J_b2, nullptr, Abuf, 4096, 256, 4096, 256, MODE_LIN);
        gemm(stream, xs,  64,  R_w1, 256,  0, R_b1, nullptr, h1R,  256,  256, 256,  64,  MODE_TANH);
        gemm(stream, h1R, 256, R_w2, 4096, 0, R_b2, nullptr, Lbuf, 4096, 256, 4096, 256, MODE_LIN);
        gemm(stream, xs,  64,  G_w1, 256,  0, G_b1, nullptr, h1G,  256,  256, 256,  64,  MODE_TANH);
        gemm(stream, h1G, 256, G_w2, 1024, 0, G_b2, nullptr, Gbuf, 1024, 256, 1024, 256, MODE_LIN);
    };

    float* xcur = xA;
    float* xnext = xB;
    const int n = 16384;
    for (int t = 0; t < 8; ++t) {
        const float* ut = u_future + t * 16;       // [b, t, :], row stride 128
        float* yh = (float*)d_out + t * 16;        // [b, t, :], row stride 128

        eval(xcur);                                // k1 eval also produces yhat
        dyn_kernel<<<256, 64, 0, stream>>>(Abuf, Lbuf, Gbuf, dHdx, ut, 128, k1, yh);

        axpy_kernel<<<64, 256, 0, stream>>>(xcur, k1, 0.5f * DT, xa, n);
        eval(xa);
        dyn_kernel<<<256, 64, 0, stream>>>(Abuf, Lbuf, Gbuf, dHdx, ut, 128, k2, nullptr);

        axpy_kernel<<<64, 256, 0, stream>>>(xcur, k2, 0.5f * DT, xa, n);
        eval(xa);
        dyn_kernel<<<256, 64, 0, stream>>>(Abuf, Lbuf, Gbuf, dHdx, ut, 128, k3, nullptr);

        axpy_kernel<<<64, 256, 0, stream>>>(xcur, k3, DT, xa, n);
        eval(xa);
        dyn_kernel<<<256, 64, 0, stream>>>(Abuf, Lbuf, Gbuf, dHdx, ut, 128, k4, nullptr);

        rk4_combine_kernel<<<64, 256, 0, stream>>>(xcur, k1, k2, k3, k4, DT / 6.0f, xnext, n);
        float* tmp = xcur; xcur = xnext; xnext = tmp;
    }
}